// BandsplitModule_23055384445079
// MI455X (gfx1250) — compile-verified
//
#include <hip/hip_runtime.h>
#include <hip/hip_bf16.h>

// Bandsplit: x(16,2,2048,1024) f32 -> mean over 32-wide bands -> per-band
// affine (K=2) -> out(16,64,1024,128) f32.
// Memory-bound (~768MB @ 23.3TB/s ~= 33us floor). Projection done exactly in
// f32 via V_WMMA_F32_16X16X4_F32. K is padded 2->4; the spare K=2 rank carries
// the bias (A[.,2]=1, B[2,.]=bias) so SRC2 is the inline-0 constant and no
// per-WMMA C broadcast is needed. x/out streamed with non-temporal hints
// (single-touch, > L2 capacity); W/b use regular loads (hot in L2).

typedef __attribute__((ext_vector_type(2))) float v2f;
typedef __attribute__((ext_vector_type(4))) float v4f;
typedef __attribute__((ext_vector_type(8))) float v8f;

#define NB 64
#define BS 32
#define EMB 128
#define TT 1024
#define FF 2048
#define CC 2

__global__ __launch_bounds__(256)
void bandsplit_wmma_kernel(const float* __restrict__ x,
                           const float* __restrict__ W,
                           const float* __restrict__ bias,
                           float* __restrict__ out)
{
    __shared__ float xm_lds[8 * 128];   // per wave: [c][64 t] band means

    const int tid  = threadIdx.x;
    const int wave = tid >> 5;
    const int lane = tid & 31;

    // 16384 waves total: gwave = ((b*64 + n)*16 + tchunk)
    const int gwave = blockIdx.x * 8 + wave;
    const int tc = gwave & 15;         // 64-wide t chunk (1024/64 = 16)
    const int n  = (gwave >> 4) & 63;  // band
    const int b  = gwave >> 10;        // batch
    const int t0 = tc * 64;

    const int c = lane >> 4;           // channel owned by this half-wave
    const int p = lane & 15;           // position within half-wave

    // ---- Phase 1: band mean. Each lane reduces a float4 column strip over
    // the 32 rows of its band. Half-wave reads 256 contiguous bytes/row.
    // Non-temporal: x is 256MB single-touch, keep it out of L2.
    const float* xp = x + ((size_t)b * CC + c) * ((size_t)FF * TT)
                        + (size_t)(n * BS) * TT
                        + (size_t)(t0 + p * 4);
    float ax = 0.f, ay = 0.f, az = 0.f, aw = 0.f;
    #pragma unroll
    for (int f = 0; f < BS; ++f) {
        const v4f v = __builtin_nontemporal_load(
            reinterpret_cast<const v4f*>(xp + (size_t)f * TT));
        ax += v.x; ay += v.y; az += v.z; aw += v.w;
    }
    const float inv = 1.0f / (float)BS;
    float* l0 = &xm_lds[wave * 128 + c * 64 + p * 4];
    l0[0] = ax * inv; l0[1] = ay * inv; l0[2] = az * inv; l0[3] = aw * inv;
    __syncthreads();

    // ---- Phase 2: B fragments. 16x4 f32 layout: lanes 0-15 hold K=0,1 in
    // the two VGPRs; lanes 16-31 hold K=2,3. K=0,1 = channels, K=2 = bias
    // rank (paired with A[.,2]=1), K=3 = zero pad.
    const float* Wn = W + n * (CC * EMB);
    const float* bn = bias + n * EMB;
    v2f Bf[8];
    #pragma unroll
    for (int e = 0; e < 8; ++e) {
        if (lane < 16) {
            Bf[e].x = Wn[e * 16 + lane];         // K=0 row (channel 0)
            Bf[e].y = Wn[EMB + e * 16 + lane];   // K=1 row (channel 1)
        } else {
            Bf[e].x = bn[e * 16 + p];            // K=2 row: bias
            Bf[e].y = 0.f;                        // K=3 pad
        }
    }

    // ---- Phase 3: 4 t-subtiles x 8 emb-subtiles of WMMA 16x16x4 f32,
    // SRC2 = inline 0 (bias carried by K=2 rank).
    float* outp = out + ((size_t)b * NB + n) * ((size_t)TT * EMB);
    const float* xm = &xm_lds[wave * 128];

    #pragma unroll
    for (int s = 0; s < 4; ++s) {
        v2f Af;
        if (lane < 16) {
            Af.x = xm[s * 16 + lane];        // K=0: channel 0, row M=lane
            Af.y = xm[64 + s * 16 + lane];   // K=1: channel 1
        } else {
            Af.x = 1.0f;                      // K=2: multiplies bias row of B
            Af.y = 0.0f;                      // K=3 pad
        }
        const int rowbase = t0 + s * 16 + ((lane >> 4) << 3);  // +8 for hi half

        #pragma unroll
        for (int e = 0; e < 8; ++e) {
            v8f Cv = {};   // lowers to inline SRC2 = 0

            // D = A x B (+0) = xm*W + bias, exact f32 path
            v8f D = __builtin_amdgcn_wmma_f32_16x16x4_f32(
                false, Af, false, Bf[e], (short)0, Cv, false, false);

            const int col = e * 16 + p;
            #pragma unroll
            for (int j = 0; j < 8; ++j)
                __builtin_nontemporal_store(
                    D[j], &outp[(size_t)(rowbase + j) * EMB + col]);
        }
    }
}

extern "C" void kernel_launch(void* const* d_in, const int* in_sizes, int n_in,
                              void* d_out, int out_size, void* d_ws, size_t ws_size,
                              hipStream_t stream) {
    const float* x    = (const float*)d_in[0];
    const float* W    = (const float*)d_in[1];
    const float* bias = (const float*)d_in[2];
    float* out        = (float*)d_out;

    // 16 batches * 64 bands * 16 t-chunks = 16384 waves; 8 waves/block.
    dim3 grid(2048), block(256);
    hipLaunchKernelGGL(bandsplit_wmma_kernel, grid, block, 0, stream,
                       x, W, bias, out);
}